// TraceableLMStep_66709432041909
// MI455X (gfx1250) — compile-verified
//
#include <hip/hip_runtime.h>
#include <math.h>

// Model dims
#define BB 2
#define DD 1024
#define HH 16
#define DHH 64
#define DFFN 4096
#define NBASE 24
#define NRES 8
#define SSEQ 512
#define SSEQ1 513

typedef __attribute__((ext_vector_type(2))) float v2f;
typedef __attribute__((ext_vector_type(8))) float v8f;
typedef unsigned int v4u __attribute__((ext_vector_type(4)));
typedef int v4i __attribute__((ext_vector_type(4)));
typedef int v8i __attribute__((ext_vector_type(8)));

// ---------------------------------------------------------------------------
// out[m][n] = sum_k x[m][k] * W[k][n], m in [0,2) padded to 16 rows for WMMA.
// Block = 8 waves; all waves share one 16-column tile, splitting K 8 ways;
// partials combined in LDS in fixed order (deterministic, no atomics).
// The x vector (2 x K fp32) is staged into LDS by the Tensor Data Mover
// (TENSOR_LOAD_TO_LDS), completion enforced with s_wait_tensorcnt.
// If rscale != 0 the epilogue performs out[idx] += rscale * result (fused
// residual); otherwise out[idx] = result.
// grid = (N/16, 1, nz); nz selects among up to 3 weight/output pairs.
// ---------------------------------------------------------------------------
__global__ __launch_bounds__(256) void gemv_wmma_kernel(
    const float* __restrict__ x,
    const float* W0, const float* W1, const float* W2,
    float* o0, float* o1, float* o2, int K, int N, float rscale) {
  __shared__ float xs[2 * DFFN];  // up to K=4096 -> 32KB
  __shared__ float red[8][2][16];
  const float* W = (blockIdx.z == 0) ? W0 : (blockIdx.z == 1) ? W1 : W2;
  float* out = (blockIdx.z == 0) ? o0 : (blockIdx.z == 1) ? o1 : o2;
  int wave = threadIdx.x >> 5;
  int lane = threadIdx.x & 31;
  int n0 = blockIdx.x * 16;

  // --- TDM: DMA x (2 rows x K cols, fp32, row stride K) into LDS -----------
  if (wave == 0) {
    unsigned lds = (unsigned)(uintptr_t)(&xs[0]);  // low 32 bits = LDS offset
    unsigned long long ga = (unsigned long long)(uintptr_t)x;
    // D# group 0: count=1 | lds_addr | global_addr[56:0] | type=2
    v4u g0 = {1u, lds, (unsigned)(ga & 0xFFFFFFFFu),
              (unsigned)((ga >> 32) & 0x01FFFFFFu) | (2u << 30)};
    // D# group 1: data_size=4B; tensor_dim0=K; tensor_dim1=2; tile_dim0=K;
    //             tile_dim1=2; tensor_dim0_stride=K
    v8i g1 = {(int)(2u << 16),
              (int)(((unsigned)K & 0xFFFFu) << 16),
              (int)((((unsigned)K >> 16) & 0xFFFFu) | (2u << 16)),
              (int)(((unsigned)K & 0xFFFFu) << 16),
              2,
              K,
              0,
              0};
    v4i gz = {0, 0, 0, 0};
#if defined(__clang_major__) && (__clang_major__ >= 23)
    v8i gz8 = {0, 0, 0, 0, 0, 0, 0, 0};
    __builtin_amdgcn_tensor_load_to_lds(g0, g1, gz, gz, gz8, 0);
#else
    __builtin_amdgcn_tensor_load_to_lds(g0, g1, gz, gz, 0);
#endif
    __builtin_amdgcn_s_wait_tensorcnt(0);
  }
  __syncthreads();

  int kPer = K >> 3;  // K/8 per wave; K in {32,1024,4096}
  int kbeg = wave * kPer;
  int kend = kbeg + kPer;
  int m = lane & 15;            // A row (batch, rows >=2 are zero) and B column
  int kk = (lane >> 4) * 2;     // K sub-index within the 4-wide WMMA step
  v8f acc0 = {0.f, 0.f, 0.f, 0.f, 0.f, 0.f, 0.f, 0.f};
  v8f acc1 = {0.f, 0.f, 0.f, 0.f, 0.f, 0.f, 0.f, 0.f};
  int k = kbeg;
  // main loop: 2 independent WMMA chains (hides WMMA->WMMA RAW latency)
  for (; k + 8 <= kend; k += 8) {
    float a0 = 0.f, a1 = 0.f, a2 = 0.f, a3 = 0.f;
    if (m < BB) {
      a0 = xs[m * K + k + kk];
      a1 = xs[m * K + k + kk + 1];
      a2 = xs[m * K + k + 4 + kk];
      a3 = xs[m * K + k + 5 + kk];
    }
    const float* wp0 = W + (size_t)(k + kk) * N + n0 + m;
    const float* wp1 = W + (size_t)(k + 4 + kk) * N + n0 + m;
    float b0 = wp0[0];
    float b1 = wp0[N];
    float b2 = wp1[0];
    float b3 = wp1[N];
    if (k + 16 <= kend) __builtin_prefetch(wp0 + 8 * (size_t)N, 0, 0);
    v2f A0 = {a0, a1};
    v2f B0 = {b0, b1};
    v2f A1 = {a2, a3};
    v2f B1 = {b2, b3};
    acc0 = __builtin_amdgcn_wmma_f32_16x16x4_f32(false, A0, false, B0,
                                                 (short)0, acc0, false, false);
    acc1 = __builtin_amdgcn_wmma_f32_16x16x4_f32(false, A1, false, B1,
                                                 (short)0, acc1, false, false);
  }
  for (; k < kend; k += 4) {
    float a0 = 0.f, a1 = 0.f;
    if (m < BB) {
      a0 = xs[m * K + k + kk];
      a1 = xs[m * K + k + kk + 1];
    }
    const float* wp = W + (size_t)(k + kk) * N + n0 + m;
    float b0 = wp[0];
    float b1 = wp[N];
    v2f A = {a0, a1};
    v2f Bv = {b0, b1};
    acc0 = __builtin_amdgcn_wmma_f32_16x16x4_f32(false, A, false, Bv, (short)0,
                                                 acc0, false, false);
  }
  // C layout: acc[j] lanes 0-15 hold row M=j; rows 0,1 are the batch.
  if (lane < 16) {
    red[wave][0][lane] = acc0[0] + acc1[0];
    red[wave][1][lane] = acc0[1] + acc1[1];
  }
  __syncthreads();
  if (threadIdx.x < 32) {
    int mm = threadIdx.x >> 4;
    int nn = threadIdx.x & 15;
    float s = 0.f;
#pragma unroll
    for (int w = 0; w < 8; ++w) s += red[w][mm][nn];
    int idx = mm * N + n0 + nn;
    if (rscale != 0.f)
      out[idx] += rscale * s;  // fused residual: h += scale * proj
    else
      out[idx] = s;
  }
}

// ---------------------------------------------------------------------------
// rmsnorm over last dim D=1024; grid = B, block = 256
// ---------------------------------------------------------------------------
__global__ void rmsnorm_kernel(const float* __restrict__ x,
                               const float* __restrict__ w,
                               float* __restrict__ out) {
  __shared__ float red[256];
  int b = blockIdx.x;
  const float* xb = x + b * DD;
  float ss = 0.f;
  for (int i = threadIdx.x; i < DD; i += 256) {
    float v = xb[i];
    ss += v * v;
  }
  red[threadIdx.x] = ss;
  __syncthreads();
  for (int o = 128; o > 0; o >>= 1) {
    if (threadIdx.x < o) red[threadIdx.x] += red[threadIdx.x + o];
    __syncthreads();
  }
  float r = rsqrtf(red[0] * (1.f / DD) + 1e-5f);
  for (int i = threadIdx.x; i < DD; i += 256) out[b * DD + i] = xb[i] * r * w[i];
}

// ---------------------------------------------------------------------------
// copy S*DH rows of a cache layer into the (S+1)*DH-strided output cache.
// grid = L*B*H, block = 256. float2 (output base is only 8B aligned).
// ---------------------------------------------------------------------------
__global__ void copy_cache_kernel(const float* __restrict__ src,
                                  float* __restrict__ dst) {
  size_t row = blockIdx.x;
  const float2* s = (const float2*)(src + row * (size_t)(SSEQ * DHH));
  float2* d = (float2*)(dst + row * (size_t)(SSEQ1 * DHH));
  for (int i = threadIdx.x; i < (SSEQ * DHH) / 2; i += 256) d[i] = s[i];
}

// ---------------------------------------------------------------------------
// RoPE on q (in place) and k, write rotated k and raw v into cache row S.
// grid = B*H, block = 64
// ---------------------------------------------------------------------------
__global__ void rope_cache_kernel(float* __restrict__ q,
                                  const float* __restrict__ k,
                                  const float* __restrict__ v,
                                  const int* __restrict__ pos,
                                  float* __restrict__ kdst,
                                  float* __restrict__ vdst) {
  int bh = blockIdx.x;
  int t = threadIdx.x;
  int b = bh / HH;
  int half = t & 31;
  // inv_freq = 10000^(-half/32) = exp(-half * ln(10000)/32)
  float ang = (float)pos[b] * __expf(-(float)half * 0.28782313662425575f);
  float sn, cs;
  sincosf(ang, &sn, &cs);
  int base = b * DD + (bh - b * HH) * DHH;
  float q1 = q[base + half], q2 = q[base + half + 32];
  float k1 = k[base + half], k2 = k[base + half + 32];
  float vv = v[base + t];
  __syncthreads();  // all reads done before in-place q write
  float qr = (t < 32) ? (q1 * cs - q2 * sn) : (q1 * sn + q2 * cs);
  float kr = (t < 32) ? (k1 * cs - k2 * sn) : (k1 * sn + k2 * cs);
  q[base + t] = qr;
  size_t coff = ((size_t)bh * SSEQ1 + SSEQ) * DHH + t;
  kdst[coff] = kr;
  vdst[coff] = vv;
}

// ---------------------------------------------------------------------------
// attention for one (b,h): scores over S+1 rows, softmax, weighted V sum.
// grid = B*H, block = 256
// ---------------------------------------------------------------------------
__global__ void attn_kernel(const float* __restrict__ q,
                            const float* __restrict__ kc,
                            const float* __restrict__ vc,
                            float* __restrict__ o) {
  __shared__ float qs[DHH];
  __shared__ float sc[SSEQ1];
  __shared__ float red[256];
  int bh = blockIdx.x;
  int t = threadIdx.x;
  int b = bh / HH;
  int h = bh - b * HH;
  if (t < DHH) qs[t] = q[b * DD + h * DHH + t];
  __syncthreads();
  const float* kbase = kc + (size_t)bh * SSEQ1 * DHH;
  for (int s = t; s < SSEQ1; s += 256) {
    const float2* kr2 = (const float2*)(kbase + s * DHH);
    float dot = 0.f;
#pragma unroll
    for (int d = 0; d < DHH / 2; ++d) {
      float2 kv = kr2[d];
      dot += qs[2 * d] * kv.x + qs[2 * d + 1] * kv.y;
    }
    sc[s] = dot * 0.125f;  // 1/sqrt(64)
  }
  __syncthreads();
  float mx = -1e30f;
  for (int s = t; s < SSEQ1; s += 256) mx = fmaxf(mx, sc[s]);
  red[t] = mx;
  __syncthreads();
  for (int o2 = 128; o2 > 0; o2 >>= 1) {
    if (t < o2) red[t] = fmaxf(red[t], red[t + o2]);
    __syncthreads();
  }
  mx = red[0];
  __syncthreads();
  float lsum = 0.f;
  for (int s = t; s < SSEQ1; s += 256) {
    float e = expf(sc[s] - mx);
    sc[s] = e;
    lsum += e;
  }
  red[t] = lsum;
  __syncthreads();
  for (int o2 = 128; o2 > 0; o2 >>= 1) {
    if (t < o2) red[t] += red[t + o2];
    __syncthreads();
  }
  float inv = 1.f / red[0];
  __syncthreads();
  int d = t & 63;
  int part = t >> 6;  // 4 partial sums over s
  const float* vbase = vc + (size_t)bh * SSEQ1 * DHH;
  float acc = 0.f;
  for (int s = part; s < SSEQ1; s += 4) acc += sc[s] * vbase[s * DHH + d];
  red[t] = acc;
  __syncthreads();
  if (part == 0)
    o[b * DD + h * DHH + d] =
        (red[d] + red[64 + d] + red[128 + d] + red[192 + d]) * inv;
}

// ---------------------------------------------------------------------------
// small elementwise kernels
// ---------------------------------------------------------------------------
__global__ void copyf_kernel(float* d, const float* s, int n) {
  int i = blockIdx.x * blockDim.x + threadIdx.x;
  if (i < n) d[i] = s[i];
}
__global__ void add_kernel(float* out, const float* a, const float* b, int n) {
  int i = blockIdx.x * blockDim.x + threadIdx.x;
  if (i < n) out[i] = a[i] + b[i];
}
__global__ void silu_mul_kernel(float* g, const float* u, int n) {
  int i = blockIdx.x * blockDim.x + threadIdx.x;
  if (i < n) {
    float x = g[i];
    g[i] = (x / (1.f + expf(-x))) * u[i];
  }
}
__global__ void bias_silu_kernel(float* x, const float* bias, int n, int dmod) {
  int i = blockIdx.x * blockDim.x + threadIdx.x;
  if (i < n) {
    float v = x[i] + bias[i % dmod];
    x[i] = v / (1.f + expf(-v));
  }
}
__global__ void fsq_quant_kernel(float* z, int n) {
  int i = blockIdx.x * blockDim.x + threadIdx.x;
  if (i < n) {
    float t = tanhf(z[i]);
    z[i] = rintf(t * 3.5f) * (1.f / 3.5f);  // FSQ_LEVELS=8 -> half=3.5
  }
}
__global__ void stop_head_kernel(const float* sp, const float* w,
                                 const float* bias, float* out) {
  __shared__ float red[256];
  int b = blockIdx.x;
  float s = 0.f;
  for (int i = threadIdx.x; i < DD; i += 256) s += sp[b * DD + i] * w[i];
  red[threadIdx.x] = s;
  __syncthreads();
  for (int o = 128; o > 0; o >>= 1) {
    if (threadIdx.x < o) red[threadIdx.x] += red[threadIdx.x + o];
    __syncthreads();
  }
  if (threadIdx.x == 0) out[b] = red[0] + bias[0];
}

// ---------------------------------------------------------------------------
extern "C" void kernel_launch(void* const* d_in, const int* in_sizes, int n_in,
                              void* d_out_v, int out_size, void* d_ws,
                              size_t ws_size, hipStream_t stream) {
  (void)in_sizes; (void)n_in; (void)out_size; (void)ws_size;
  const float* embed = (const float*)d_in[0];
  const int* position = (const int*)d_in[1];
  const float* base_k = (const float*)d_in[2];
  const float* base_v = (const float*)d_in[3];
  const float* res_k = (const float*)d_in[4];
  const float* res_v = (const float*)d_in[5];
  const float* base_ln1 = (const float*)d_in[6];
  const float* base_ln2 = (const float*)d_in[7];
  const float* base_wq = (const float*)d_in[8];
  const float* base_wk = (const float*)d_in[9];
  const float* base_wv = (const float*)d_in[10];
  const float* base_wo = (const float*)d_in[11];
  const float* base_wg = (const float*)d_in[12];
  const float* base_wu = (const float*)d_in[13];
  const float* base_wd = (const float*)d_in[14];
  const float* base_norm = (const float*)d_in[15];
  const float* res_ln1 = (const float*)d_in[16];
  const float* res_ln2 = (const float*)d_in[17];
  const float* res_wq = (const float*)d_in[18];
  const float* res_wk = (const float*)d_in[19];
  const float* res_wv = (const float*)d_in[20];
  const float* res_wo = (const float*)d_in[21];
  const float* res_wg = (const float*)d_in[22];
  const float* res_wu = (const float*)d_in[23];
  const float* res_wd = (const float*)d_in[24];
  const float* res_norm = (const float*)d_in[25];
  const float* fsq_in = (const float*)d_in[26];
  const float* fsq_out = (const float*)d_in[27];
  const float* stop_proj_w = (const float*)d_in[28];
  const float* stop_proj_b = (const float*)d_in[29];
  const float* stop_head_w = (const float*)d_in[30];
  const float* stop_head_b = (const float*)d_in[31];

  // outputs: lm_fsq(2048), rh(2048), stop(2), bnk, bnv, rnk, rnv
  float* out = (float*)d_out_v;
  float* lm_fsq_o = out;
  float* rh_o = out + 2048;
  float* stop_o = out + 4096;
  const size_t baseCacheN = (size_t)NBASE * BB * HH * SSEQ1 * DHH;
  const size_t resCacheN = (size_t)NRES * BB * HH * SSEQ1 * DHH;
  float* bnk = out + 4098;
  float* bnv = bnk + baseCacheN;
  float* rnk = bnv + baseCacheN;
  float* rnv = rnk + resCacheN;

  // workspace layout (floats)
  float* ws = (float*)d_ws;
  float* h = ws;              // 2048
  float* hn = ws + 2048;      // 2048
  float* q = ws + 4096;       // 2048
  float* kbuf = ws + 6144;    // 2048
  float* vbuf = ws + 8192;    // 2048
  float* oatt = ws + 10240;   // 2048
  float* g = ws + 12288;      // 8192
  float* u = ws + 20480;      // 8192
  float* lmh = ws + 28672;    // 2048
  float* zf = ws + 30720;     // 64
  float* sp = ws + 30784;     // 2048

  // 1) copy existing KV caches into output (rows 0..S-1 of the S+1 axis)
  copy_cache_kernel<<<NBASE * BB * HH, 256, 0, stream>>>(base_k, bnk);
  copy_cache_kernel<<<NBASE * BB * HH, 256, 0, stream>>>(base_v, bnv);
  copy_cache_kernel<<<NRES * BB * HH, 256, 0, stream>>>(res_k, rnk);
  copy_cache_kernel<<<NRES * BB * HH, 256, 0, stream>>>(res_v, rnv);

  auto run_layers = [&](int L, const float* ln1, const float* ln2,
                        const float* wq, const float* wk, const float* wv,
                        const float* wo, const float* wg, const float* wu,
                        const float* wd, float* nk, float* nv, float scale) {
    const size_t ldd = (size_t)DD * DD;
    const size_t lff = (size_t)DD * DFFN;
    const size_t lcache = (size_t)BB * HH * SSEQ1 * DHH;
    for (int l = 0; l < L; ++l) {
      const float* wq_l = wq + (size_t)l * ldd;
      const float* wk_l = wk + (size_t)l * ldd;
      const float* wv_l = wv + (size_t)l * ldd;
      const float* wo_l = wo + (size_t)l * ldd;
      const float* wg_l = wg + (size_t)l * lff;
      const float* wu_l = wu + (size_t)l * lff;
      const float* wd_l = wd + (size_t)l * lff;
      float* nk_l = nk + (size_t)l * lcache;
      float* nv_l = nv + (size_t)l * lcache;

      rmsnorm_kernel<<<BB, 256, 0, stream>>>(h, ln1 + (size_t)l * DD, hn);
      // fused QKV GEMV (WMMA + TDM staging)
      gemv_wmma_kernel<<<dim3(DD / 16, 1, 3), 256, 0, stream>>>(
          hn, wq_l, wk_l, wv_l, q, kbuf, vbuf, DD, DD, 0.f);
      rope_cache_kernel<<<BB * HH, 64, 0, stream>>>(q, kbuf, vbuf, position,
                                                    nk_l, nv_l);
      attn_kernel<<<BB * HH, 256, 0, stream>>>(q, nk_l, nv_l, oatt);
      // o-projection with fused residual: h += scale * (oatt @ wo)
      gemv_wmma_kernel<<<dim3(DD / 16, 1, 1), 256, 0, stream>>>(
          oatt, wo_l, wo_l, wo_l, h, h, h, DD, DD, scale);

      rmsnorm_kernel<<<BB, 256, 0, stream>>>(h, ln2 + (size_t)l * DD, hn);
      // fused gate/up GEMV
      gemv_wmma_kernel<<<dim3(DFFN / 16, 1, 2), 256, 0, stream>>>(
          hn, wg_l, wu_l, wu_l, g, u, u, DD, DFFN, 0.f);
      silu_mul_kernel<<<32, 256, 0, stream>>>(g, u, BB * DFFN);
      // down-projection with fused residual: h += scale * (m @ wd)
      gemv_wmma_kernel<<<dim3(DD / 16, 1, 1), 256, 0, stream>>>(
          g, wd_l, wd_l, wd_l, h, h, h, DFFN, DD, scale);
    }
  };

  // 2) base LM
  copyf_kernel<<<8, 256, 0, stream>>>(h, embed, BB * DD);
  run_layers(NBASE, base_ln1, base_ln2, base_wq, base_wk, base_wv, base_wo,
             base_wg, base_wu, base_wd, bnk, bnv, 1.4f / sqrtf((float)NBASE));

  // 3) lm_hidden, FSQ, stop head
  rmsnorm_kernel<<<BB, 256, 0, stream>>>(h, base_norm, lmh);
  gemv_wmma_kernel<<<dim3(32 / 16, 1, 1), 256, 0, stream>>>(
      lmh, fsq_in, fsq_in, fsq_in, zf, zf, zf, DD, 32, 0.f);
  fsq_quant_kernel<<<1, 64, 0, stream>>>(zf, BB * 32);
  gemv_wmma_kernel<<<dim3(DD / 16, 1, 1), 256, 0, stream>>>(
      zf, fsq_out, fsq_out, fsq_out, lm_fsq_o, lm_fsq_o, lm_fsq_o, 32, DD, 0.f);
  gemv_wmma_kernel<<<dim3(DD / 16, 1, 1), 256, 0, stream>>>(
      lmh, stop_proj_w, stop_proj_w, stop_proj_w, sp, sp, sp, DD, DD, 0.f);
  bias_silu_kernel<<<8, 256, 0, stream>>>(sp, stop_proj_b, BB * DD, DD);
  stop_head_kernel<<<BB, 256, 0, stream>>>(sp, stop_head_w, stop_head_b,
                                           stop_o);

  // 4) residual LM: res_in = lm_fsq + embed
  add_kernel<<<8, 256, 0, stream>>>(h, lm_fsq_o, embed, BB * DD);
  run_layers(NRES, res_ln1, res_ln2, res_wq, res_wk, res_wv, res_wo, res_wg,
             res_wu, res_wd, rnk, rnv, 1.4f / sqrtf((float)NRES));
  rmsnorm_kernel<<<BB, 256, 0, stream>>>(h, res_norm, rh_o);
}